// NearestNeighborsLayer_70677981823086
// MI455X (gfx1250) — compile-verified
//
#include <hip/hip_runtime.h>

typedef __attribute__((ext_vector_type(2))) float v2f;
typedef __attribute__((ext_vector_type(8))) float v8f;

#define BATCH   8
#define NPTS    16384   // source cloud size (xyz1)
#define MQRY    8192    // query cloud size (xyz2)
#define KNN     16
#define CHUNK   2048    // points staged in LDS per iteration
#define TILES   (CHUNK / 16)
#define WAVES   8       // waves per block (256 threads, wave32)

__global__ __launch_bounds__(256)
void knn_wmma_kernel(const float* __restrict__ xyz1,
                     const float* __restrict__ xyz2,
                     int* __restrict__ out)
{
    // Points staged directly in WMMA A-operand layout:
    //   a_lds[t][lane] for tile t of 16 points:
    //     lanes 0-15  -> (x, y)        of point t*16 + lane   (K0,K1)
    //     lanes 16-31 -> (z, |p|^2)    of point t*16 + lane-16 (K2,K3)
    __shared__ v2f   a_lds[TILES][32];         // 32 KB
    __shared__ float mrg_d[WAVES][32][KNN];    // 16 KB
    __shared__ int   mrg_i[WAVES][32][KNN];    // 16 KB

    const int lane  = threadIdx.x & 31;
    const int wave  = threadIdx.x >> 5;
    const int batch = blockIdx.x >> 6;                  // 64 blocks per batch
    const int qTile = (blockIdx.x & 63) * WAVES + wave; // 0..511
    const int qcol  = lane & 15;
    const int half  = lane >> 4;
    const int q     = qTile * 16 + qcol;

    // B operand (queries), loop-invariant:
    //   lanes 0-15  : K0,K1 = (-2qx, -2qy) for column N=lane
    //   lanes 16-31 : K2,K3 = (-2qz,  1  ) for column N=lane-16
    const float* qp = xyz2 + ((size_t)batch * MQRY + q) * 3;
    float qx = qp[0], qy = qp[1], qz = qp[2];
    v2f bop;
    bop.x = half ? (-2.0f * qz) : (-2.0f * qx);
    bop.y = half ? 1.0f         : (-2.0f * qy);

    // Per-lane sorted top-16 (ascending), register resident.
    float dk[KNN];
    int   ik[KNN];
#pragma unroll
    for (int i = 0; i < KNN; ++i) { dk[i] = 3.402823466e38f; ik[i] = 0; }

    const float* pbatch = xyz1 + (size_t)batch * NPTS * 3;

    for (int c = 0; c < NPTS / CHUNK; ++c) {
        __syncthreads();
        // Cooperative stage: 256 threads x 8 points, pre-swizzled to A layout.
        for (int i = threadIdx.x; i < CHUNK; i += 256) {
            const float* pp = pbatch + (size_t)(c * CHUNK + i) * 3;
            float x = pp[0], y = pp[1], z = pp[2];
            int t = i >> 4, j = i & 15;
            v2f lo; lo.x = x; lo.y = y;
            v2f hi; hi.x = z; hi.y = x * x + y * y + z * z;
            a_lds[t][j]      = lo;
            a_lds[t][j + 16] = hi;
        }
        __syncthreads();

        // 2 tiles per iteration: batch the ds_loads, issue WMMAs back-to-back.
        for (int t = 0; t < TILES; t += 2) {
            v2f a0 = a_lds[t][lane];
            v2f a1 = a_lds[t + 1][lane];

            v8f cz = {0.f, 0.f, 0.f, 0.f, 0.f, 0.f, 0.f, 0.f};
            // D[M=point][N=query] = -2 q.p + |p|^2 (row-constant |q|^2 omitted:
            // it cannot change per-query top-k ordering)
            v8f d0 = __builtin_amdgcn_wmma_f32_16x16x4_f32(
                false, a0, false, bop, (short)0, cz, false, false);
            v8f d1 = __builtin_amdgcn_wmma_f32_16x16x4_f32(
                false, a1, false, bop, (short)0, cz, false, false);

            const int pb0 = c * CHUNK + t * 16 + half * 8;
            const int pb1 = pb0 + 16;

#pragma unroll
            for (int r = 0; r < 8; ++r) {
                float dist = d0[r];
                int   pidx = pb0 + r;
                if (dist < dk[KNN - 1]) {
                    float cd = dist; int ci = pidx;
#pragma unroll
                    for (int j = 0; j < KNN; ++j) {
                        bool swp = cd < dk[j];
                        float td = dk[j]; int ti = ik[j];
                        dk[j] = swp ? cd : dk[j];
                        ik[j] = swp ? ci : ik[j];
                        cd = swp ? td : cd;
                        ci = swp ? ti : ci;
                    }
                }
            }
#pragma unroll
            for (int r = 0; r < 8; ++r) {
                float dist = d1[r];
                int   pidx = pb1 + r;
                if (dist < dk[KNN - 1]) {
                    float cd = dist; int ci = pidx;
#pragma unroll
                    for (int j = 0; j < KNN; ++j) {
                        bool swp = cd < dk[j];
                        float td = dk[j]; int ti = ik[j];
                        dk[j] = swp ? cd : dk[j];
                        ik[j] = swp ? ci : ik[j];
                        cd = swp ? td : cd;
                        ci = swp ? ti : ci;
                    }
                }
            }
        }
    }

    // Merge the two sorted half-lists (lanes l and l+16 own the same query).
    __syncthreads();
#pragma unroll
    for (int k = 0; k < KNN; ++k) {
        mrg_d[wave][lane][k] = dk[k];
        mrg_i[wave][lane][k] = ik[k];
    }
    __syncthreads();

    if (half == 0) {
        const float* da  = mrg_d[wave][qcol];
        const float* db  = mrg_d[wave][qcol + 16];
        const int*   ia_ = mrg_i[wave][qcol];
        const int*   ib_ = mrg_i[wave][qcol + 16];
        int* o = out + ((size_t)batch * MQRY + q) * KNN;
        int ia = 0, ib = 0;
#pragma unroll
        for (int k = 0; k < KNN; ++k) {
            float va = da[ia], vb = db[ib];
            if (va <= vb) { o[k] = ia_[ia]; ++ia; }
            else          { o[k] = ib_[ib]; ++ib; }
        }
    }
}

extern "C" void kernel_launch(void* const* d_in, const int* in_sizes, int n_in,
                              void* d_out, int out_size, void* d_ws, size_t ws_size,
                              hipStream_t stream) {
    (void)in_sizes; (void)n_in; (void)out_size; (void)d_ws; (void)ws_size;
    const float* xyz1 = (const float*)d_in[0];
    const float* xyz2 = (const float*)d_in[1];
    int* out = (int*)d_out;

    dim3 grid(BATCH * (MQRY / (16 * WAVES)));  // 512 blocks
    dim3 block(256);                           // 8 waves of 32
    hipLaunchKernelGGL(knn_wmma_kernel, grid, block, 0, stream, xyz1, xyz2, out);
}